// ETP_26740466385025
// MI455X (gfx1250) — compile-verified
//
#include <hip/hip_runtime.h>
#include <hip/hip_bf16.h>

// Entropic OT / Sinkhorn for x:(1024,512), y:(16384,512) fp32.
//
// Roofline: GEMM is 17.2 GFLOP (trivial); the cost is data movement.
//   - M/K matrix is 64 MB -> fits in 192 MB L2, so Sinkhorn iterations
//     (2 matvecs = 128 MB/iter) run at L2 bandwidth.
//   - d_out P write = 64 MB @ 23.3 TB/s ~ 2.7 us.
// Matrix ops use native fp32 WMMA (v_wmma_f32_16x16x4_f32) for exact
// squared distances (f16 would corrupt exp(-0.1*(M-min))).

#define GAMMA   0.1f
#define EPSERR  0.005f
#define MAXIT   1000
#define TINYV   1e-30f

#define NROWS   1024
#define NCOLS   16384
#define KDIM    512

#define NB      64      // persistent-kernel blocks (must all be resident);
                        // 64*256 threads = exactly one thread per column in phase B
#define BS      256

typedef __attribute__((ext_vector_type(2))) float v2f;
typedef __attribute__((ext_vector_type(8))) float v8f;

// ---------------- workspace layout (floats) ----------------
// [0,1024)        rowmin bit-patterns (uint)
// [1024,2048)     xn  (||x_i||^2)
// [2048,18432)    yn  (||y_j||^2)
// [18432,19456)   u
// [19456,35840)   v
// [35840,35842)   errbuf[2]
// [35842,35844)   sync: cnt, gen
#define WS_ROWMIN 0
#define WS_XN     1024
#define WS_YN     2048
#define WS_U      18432
#define WS_V      19456
#define WS_ERR    35840
#define WS_SYNC   35842

__global__ void k_init(float* out0, unsigned* rowmin_bits, float* u, float* v,
                       float* errbuf, unsigned* syncw) {
  int gid = blockIdx.x * blockDim.x + threadIdx.x;
  if (gid == 0) {
    out0[0] = 0.0f;
    errbuf[0] = 0.0f; errbuf[1] = 0.0f;
    syncw[0] = 0u; syncw[1] = 0u;
  }
  if (gid < NROWS) { rowmin_bits[gid] = 0x7F800000u; u[gid] = 1.0f; }
  if (gid < NCOLS) v[gid] = 1.0f;
}

// one block per row (x rows then y rows): squared L2 norm
__global__ __launch_bounds__(128)
void k_norms(const float* __restrict__ x, const float* __restrict__ y,
             float* __restrict__ xn, float* __restrict__ yn) {
  __shared__ float red[128];
  int r = blockIdx.x;
  const float* p = (r < NROWS) ? (x + (size_t)r * KDIM)
                               : (y + (size_t)(r - NROWS) * KDIM);
  float s = 0.0f;
  for (int k = threadIdx.x; k < KDIM; k += 128) { float t = p[k]; s += t * t; }
  red[threadIdx.x] = s; __syncthreads();
  for (int off = 64; off > 0; off >>= 1) {
    if (threadIdx.x < off) red[threadIdx.x] += red[threadIdx.x + off];
    __syncthreads();
  }
  if (threadIdx.x == 0) {
    if (r < NROWS) xn[r] = red[0]; else yn[r - NROWS] = red[0];
  }
}

// WMMA fp32 GEMM: each wave computes a 16(M) x 64(N) strip of M = xn+yn-2*x.yT
// Also reduces per-row min into rowmin_bits (ordered-int atomicMin, M >= 0).
__global__ __launch_bounds__(256)
void k_gemm_M(const float* __restrict__ x, const float* __restrict__ y,
              const float* __restrict__ xn, const float* __restrict__ yn,
              float* __restrict__ Mout, unsigned* __restrict__ rowmin_bits) {
  const int lane = threadIdx.x & 31;
  const int wave = threadIdx.x >> 5;
  const int w    = blockIdx.x * 8 + wave;   // 0..16383
  const int rt   = w >> 8;                  // row tile 0..63
  const int cs   = w & 255;                 // col strip 0..255
  const int row0 = rt * 16;
  const int col0 = cs * 64;
  const int half = lane >> 4;               // K sub-pair select
  const int lr   = lane & 15;               // M (A) or N (B) index

  const float* xrow  = x + (size_t)(row0 + lr) * KDIM + half * 2;
  const float* yrow0 = y + (size_t)(col0 +  0 + lr) * KDIM + half * 2;
  const float* yrow1 = y + (size_t)(col0 + 16 + lr) * KDIM + half * 2;
  const float* yrow2 = y + (size_t)(col0 + 32 + lr) * KDIM + half * 2;
  const float* yrow3 = y + (size_t)(col0 + 48 + lr) * KDIM + half * 2;

  v8f acc0 = {}, acc1 = {}, acc2 = {}, acc3 = {};
  for (int kk = 0; kk < KDIM; kk += 4) {
    v2f a  = *(const v2f*)(xrow  + kk);
    v2f b0 = *(const v2f*)(yrow0 + kk);
    v2f b1 = *(const v2f*)(yrow1 + kk);
    v2f b2 = *(const v2f*)(yrow2 + kk);
    v2f b3 = *(const v2f*)(yrow3 + kk);
    acc0 = __builtin_amdgcn_wmma_f32_16x16x4_f32(false, a, false, b0, (short)0, acc0, false, false);
    acc1 = __builtin_amdgcn_wmma_f32_16x16x4_f32(false, a, false, b1, (short)0, acc1, false, false);
    acc2 = __builtin_amdgcn_wmma_f32_16x16x4_f32(false, a, false, b2, (short)0, acc2, false, false);
    acc3 = __builtin_amdgcn_wmma_f32_16x16x4_f32(false, a, false, b3, (short)0, acc3, false, false);
  }

  const float yn0 = yn[col0 +  0 + lr];
  const float yn1 = yn[col0 + 16 + lr];
  const float yn2 = yn[col0 + 32 + lr];
  const float yn3 = yn[col0 + 48 + lr];

  for (int vg = 0; vg < 8; ++vg) {
    const int row = row0 + vg + 8 * half;   // C/D layout: M = vg + 8*(lane/16)
    const float xnr = xn[row];
    float m0 = xnr + yn0 - 2.0f * acc0[vg];
    float m1 = xnr + yn1 - 2.0f * acc1[vg];
    float m2 = xnr + yn2 - 2.0f * acc2[vg];
    float m3 = xnr + yn3 - 2.0f * acc3[vg];
    size_t base = (size_t)row * NCOLS + col0 + lr;
    Mout[base +  0] = m0;
    Mout[base + 16] = m1;
    Mout[base + 32] = m2;
    Mout[base + 48] = m3;
    float mn = fminf(fminf(m0, m1), fminf(m2, m3));
    mn = fminf(mn, __shfl_xor(mn, 1, 32));
    mn = fminf(mn, __shfl_xor(mn, 2, 32));
    mn = fminf(mn, __shfl_xor(mn, 4, 32));
    mn = fminf(mn, __shfl_xor(mn, 8, 32));
    if (lr == 0) atomicMin(&rowmin_bits[row], __float_as_uint(mn));
  }
}

// K = exp(-gamma*(M - rowmin)) in place
__global__ __launch_bounds__(256)
void k_expK(float* __restrict__ P, const unsigned* __restrict__ rowmin_bits) {
  size_t stride = (size_t)gridDim.x * blockDim.x;
  size_t total  = (size_t)NROWS * NCOLS;
  for (size_t idx = (size_t)blockIdx.x * blockDim.x + threadIdx.x;
       idx < total; idx += stride) {
    int row  = (int)(idx >> 14);
    float rm = __uint_as_float(rowmin_bits[row]);
    P[idx]   = __expf(-GAMMA * (P[idx] - rm));
  }
}

__device__ inline void gsync(unsigned* cnt, unsigned* gen) {
  __threadfence();
  __syncthreads();
  if (threadIdx.x == 0) {
    unsigned g = __hip_atomic_load(gen, __ATOMIC_RELAXED, __HIP_MEMORY_SCOPE_AGENT);
    if (atomicAdd(cnt, 1u) == NB - 1) {
      *cnt = 0u;
      __threadfence();
      atomicAdd(gen, 1u);
    } else {
      while (__hip_atomic_load(gen, __ATOMIC_RELAXED, __HIP_MEMORY_SCOPE_AGENT) == g)
        __builtin_amdgcn_s_sleep(1);
    }
  }
  __threadfence();
  __syncthreads();
}

// Persistent Sinkhorn on scaling vectors: P = diag(u) K diag(v).
// Per iter: u_i = r*u_i/max(u_i * (Kv)_i, TINY);  t_j = (K^T u)_j;
// err = sum_j |v_j t_j - c|;  if err > EPS: v_j = c*v_j/max(v_j t_j, TINY).
// K (64MB) stays L2-resident, so the two matvecs run at L2 bandwidth.
__global__ __launch_bounds__(BS)
void k_sinkhorn(const float* __restrict__ K, float* __restrict__ u,
                float* __restrict__ v, float* errbuf, unsigned* syncw) {
  __shared__ float red[BS];
  __shared__ float ush[NROWS];
  const float rmass = 1.0f / NROWS;
  const float cmass = 1.0f / NCOLS;
  const int tid = threadIdx.x;
  const int gid = blockIdx.x * BS + tid;   // NB*BS == NCOLS: one col per thread

  for (int it = 0; it < MAXIT; ++it) {
    // ---- phase A: s_i = sum_j K_ij v_j ; update u_i ----
    for (int row = blockIdx.x; row < NROWS; row += NB) {
      const float* Krow = K + (size_t)row * NCOLS;
      float s = 0.0f;
      for (int j = tid; j < NCOLS; j += BS)   // b32 loads: K is 4B-aligned only
        s += Krow[j] * v[j];
      red[tid] = s; __syncthreads();
      for (int off = BS / 2; off > 0; off >>= 1) {
        if (tid < off) red[tid] += red[tid + off];
        __syncthreads();
      }
      if (tid == 0) {
        float uo = u[row];
        u[row] = rmass * uo / fmaxf(uo * red[0], TINYV);
      }
      __syncthreads();
    }
    gsync(&syncw[0], &syncw[1]);

    // stage u in LDS (read 1024x per thread below)
    for (int i = tid; i < NROWS; i += BS) ush[i] = u[i];
    __syncthreads();

    // ---- phase B: t_j = sum_i K_ij u_i ; error ----
    const int j = gid;                       // always < NCOLS with NB=64
    const float* Kc = K + j;
    float t = 0.0f;
    for (int i = 0; i < NROWS; i += 4) {
      t += Kc[(size_t)(i + 0) * NCOLS] * ush[i + 0]
         + Kc[(size_t)(i + 1) * NCOLS] * ush[i + 1]
         + Kc[(size_t)(i + 2) * NCOLS] * ush[i + 2]
         + Kc[(size_t)(i + 3) * NCOLS] * ush[i + 3];
    }
    float vj   = v[j];
    float beta = vj * t;
    float errLocal = fabsf(beta - cmass);

    red[tid] = errLocal; __syncthreads();
    for (int off = BS / 2; off > 0; off >>= 1) {
      if (tid < off) red[tid] += red[tid + off];
      __syncthreads();
    }
    const int p = it & 1;
    if (tid == 0) atomicAdd(&errbuf[p], red[0]);
    gsync(&syncw[0], &syncw[1]);

    float err = __hip_atomic_load(&errbuf[p], __ATOMIC_RELAXED, __HIP_MEMORY_SCOPE_AGENT);
    const bool converged = (err <= EPSERR);
    if (!converged)
      v[j] = cmass * vj / fmaxf(beta, TINYV); // matches reference P2 branch
    if (gid == 0) errbuf[p ^ 1] = 0.0f;       // recycle other slot
    gsync(&syncw[0], &syncw[1]);
    if (converged) break;                     // reference returns P1 (old v)
  }
}

// Recompute M via WMMA, write P = u_i * K_ij * v_j in place, reduce sum(P*M).
__global__ __launch_bounds__(256)
void k_finalize(const float* __restrict__ x, const float* __restrict__ y,
                const float* __restrict__ xn, const float* __restrict__ yn,
                const float* __restrict__ u, const float* __restrict__ v,
                float* __restrict__ P, float* __restrict__ loss) {
  const int lane = threadIdx.x & 31;
  const int wave = threadIdx.x >> 5;
  const int w    = blockIdx.x * 8 + wave;
  const int rt   = w >> 8;
  const int cs   = w & 255;
  const int row0 = rt * 16;
  const int col0 = cs * 64;
  const int half = lane >> 4;
  const int lr   = lane & 15;

  const float* xrow  = x + (size_t)(row0 + lr) * KDIM + half * 2;
  const float* yrow0 = y + (size_t)(col0 +  0 + lr) * KDIM + half * 2;
  const float* yrow1 = y + (size_t)(col0 + 16 + lr) * KDIM + half * 2;
  const float* yrow2 = y + (size_t)(col0 + 32 + lr) * KDIM + half * 2;
  const float* yrow3 = y + (size_t)(col0 + 48 + lr) * KDIM + half * 2;

  v8f acc0 = {}, acc1 = {}, acc2 = {}, acc3 = {};
  for (int kk = 0; kk < KDIM; kk += 4) {
    v2f a  = *(const v2f*)(xrow  + kk);
    v2f b0 = *(const v2f*)(yrow0 + kk);
    v2f b1 = *(const v2f*)(yrow1 + kk);
    v2f b2 = *(const v2f*)(yrow2 + kk);
    v2f b3 = *(const v2f*)(yrow3 + kk);
    acc0 = __builtin_amdgcn_wmma_f32_16x16x4_f32(false, a, false, b0, (short)0, acc0, false, false);
    acc1 = __builtin_amdgcn_wmma_f32_16x16x4_f32(false, a, false, b1, (short)0, acc1, false, false);
    acc2 = __builtin_amdgcn_wmma_f32_16x16x4_f32(false, a, false, b2, (short)0, acc2, false, false);
    acc3 = __builtin_amdgcn_wmma_f32_16x16x4_f32(false, a, false, b3, (short)0, acc3, false, false);
  }

  const float yn0 = yn[col0 +  0 + lr], v0 = v[col0 +  0 + lr];
  const float yn1 = yn[col0 + 16 + lr], v1 = v[col0 + 16 + lr];
  const float yn2 = yn[col0 + 32 + lr], v2 = v[col0 + 32 + lr];
  const float yn3 = yn[col0 + 48 + lr], v3 = v[col0 + 48 + lr];

  float lossLocal = 0.0f;
  for (int vg = 0; vg < 8; ++vg) {
    const int row = row0 + vg + 8 * half;
    const float xnr = xn[row];
    const float ur  = u[row];
    size_t base = (size_t)row * NCOLS + col0 + lr;
    float m0 = xnr + yn0 - 2.0f * acc0[vg];
    float m1 = xnr + yn1 - 2.0f * acc1[vg];
    float m2 = xnr + yn2 - 2.0f * acc2[vg];
    float m3 = xnr + yn3 - 2.0f * acc3[vg];
    float p0 = ur * P[base +  0] * v0;
    float p1 = ur * P[base + 16] * v1;
    float p2 = ur * P[base + 32] * v2;
    float p3 = ur * P[base + 48] * v3;
    P[base +  0] = p0;
    P[base + 16] = p1;
    P[base + 32] = p2;
    P[base + 48] = p3;
    lossLocal += p0 * m0 + p1 * m1 + p2 * m2 + p3 * m3;
  }
  for (int off = 16; off > 0; off >>= 1)
    lossLocal += __shfl_down(lossLocal, off, 32);
  __shared__ float wred[8];
  if (lane == 0) wred[wave] = lossLocal;
  __syncthreads();
  if (threadIdx.x == 0) {
    float s = 0.0f;
    for (int i = 0; i < 8; ++i) s += wred[i];
    atomicAdd(loss, s);
  }
}

extern "C" void kernel_launch(void* const* d_in, const int* in_sizes, int n_in,
                              void* d_out, int out_size, void* d_ws, size_t ws_size,
                              hipStream_t stream) {
  const float* x = (const float*)d_in[0];   // (1024, 512)
  const float* y = (const float*)d_in[1];   // (16384, 512)
  float* out = (float*)d_out;               // [0]=loss, [1..]=P (1024x16384)
  float* ws  = (float*)d_ws;

  unsigned* rowmin_bits = (unsigned*)(ws + WS_ROWMIN);
  float*    xn    = ws + WS_XN;
  float*    yn    = ws + WS_YN;
  float*    u     = ws + WS_U;
  float*    v     = ws + WS_V;
  float*    errb  = ws + WS_ERR;
  unsigned* syncw = (unsigned*)(ws + WS_SYNC);
  float*    P     = out + 1;                // K during iterations, P at the end

  k_init    <<<64, 256, 0, stream>>>(out, rowmin_bits, u, v, errb, syncw);
  k_norms   <<<NROWS + NCOLS, 128, 0, stream>>>(x, y, xn, yn);
  k_gemm_M  <<<2048, 256, 0, stream>>>(x, y, xn, yn, P, rowmin_bits);
  k_expK    <<<2048, 256, 0, stream>>>(P, rowmin_bits);
  k_sinkhorn<<<NB, BS, 0, stream>>>(P, u, v, errb, syncw);
  k_finalize<<<2048, 256, 0, stream>>>(x, y, xn, yn, u, v, P, out);
}